// ToRGB_87325275062852
// MI455X (gfx1250) — compile-verified
//
#include <hip/hip_runtime.h>
#include <math.h>

typedef __attribute__((ext_vector_type(2))) float v2f;
typedef __attribute__((ext_vector_type(8))) float v8f;

#define CIN      512
#define CO       3
#define H        256
#define W        256
#define HW       65536
#define NB       8
#define KC       16            // channels per LDS chunk
#define NCHUNK   (CIN / KC)    // 32
#define PIX_BLK  64            // pixels per workgroup (4 waves x 16)
#define INV_SQRT512 0.04419417382415922f

// ---- CDNA5 async global->LDS copy (gfx1250), tracked by ASYNCcnt ----
// Non-temporal: x is a 1.07GB single-pass stream; keep it out of L2 residency.
__device__ __forceinline__ void async_copy_b128_nt(unsigned lds_byte_addr, const float* g) {
    asm volatile("global_load_async_to_lds_b128 %0, %1, off th:TH_LOAD_NT"
                 :: "v"(lds_byte_addr), "v"((unsigned long long)(size_t)g)
                 : "memory");
}
#define WAIT_ASYNC_LE2() asm volatile("s_wait_asynccnt 0x2" ::: "memory")
#define WAIT_ASYNC_LE0() asm volatile("s_wait_asynccnt 0x0" ::: "memory")

// =====================================================================
// K1: 1x1 conv (512->3) via V_WMMA_F32_16X16X4_F32, async-LDS pipelined.
// Writes y = conv(x)*scale + conv_b + out_bias into yout (= d_out).
// =====================================================================
__global__ void __launch_bounds__(128)
conv_wmma(const float* __restrict__ xg, const float* __restrict__ conv_w,
          const float* __restrict__ conv_b, const float* __restrict__ out_bias,
          float* __restrict__ yout)
{
    __shared__ float xbuf[2][KC * PIX_BLK];   // 2 x 4KB double buffer
    __shared__ float wlds[CO * CIN];          // 6KB scaled weights [o][c]

    const int t    = threadIdx.x;
    const int wave = t >> 5;
    const int lane = t & 31;
    const int m     = lane & 15;   // A: pixel row M / B: column N
    const int khalf = lane >> 4;   // 0 -> K{0,1}, 1 -> K{2,3}
    const int nw    = (m < 3) ? m : 0;
    const float nvalid = (m < 3) ? 1.0f : 0.0f;

    // block -> (b, h, w0)
    const int pixBase = blockIdx.x * PIX_BLK;
    const int b   = pixBase >> 16;
    const int rem = pixBase & (HW - 1);
    const int h   = rem >> 8;
    const int w0  = rem & (W - 1);

    // stage scaled weights once per block: 384 float4s, 3 per thread,
    // issued as independent b128 loads (no per-iteration load/wait chain)
    {
        const float4* cw4 = (const float4*)conv_w;
        #pragma unroll
        for (int i = 0; i < 3; ++i) {
            const int idx = t + i * 128;               // < 384
            float4 v = cw4[idx];
            v.x *= INV_SQRT512; v.y *= INV_SQRT512;
            v.z *= INV_SQRT512; v.w *= INV_SQRT512;
            *(float4*)(&wlds[idx * 4]) = v;
        }
    }

    // producer: copy chunk ci (16 ch x 64 pix) into xbuf[ci&1]
    const size_t pixOff = (size_t)h * W + w0;
    const int r    = t >> 3;        // channel row within chunk: 0..15
    const int seg0 = t & 7;         // 16B segment 0..7 (halves give 0..15)
    auto issue_chunk = [&](int ci) {
        float* dst = xbuf[ci & 1];
        const size_t chan0 = (size_t)(b * CIN + ci * KC + r) << 16; // *HW
        #pragma unroll
        for (int half = 0; half < 2; ++half) {
            const int seg = seg0 + half * 8;
            unsigned lb = (unsigned)(size_t)(dst + r * PIX_BLK + seg * 4);
            async_copy_b128_nt(lb, xg + chan0 + pixOff + (seg << 2));
        }
    };

    v8f acc = {};
    issue_chunk(0);

    for (int ci = 0; ci < NCHUNK; ++ci) {
        if (ci + 1 < NCHUNK) { issue_chunk(ci + 1); WAIT_ASYNC_LE2(); }
        else                 { WAIT_ASYNC_LE0(); }
        __syncthreads();                       // chunk ci visible to all waves

        const float* xb = xbuf[ci & 1];
        #pragma unroll
        for (int j = 0; j < 4; ++j) {          // 4 K-steps of 4 channels
            const int cl = j * 4 + khalf * 2;  // local channel of K0 for this lane-half
            v2f a; a[0] = xb[(cl + 0) * PIX_BLK + (wave << 4) + m];
                   a[1] = xb[(cl + 1) * PIX_BLK + (wave << 4) + m];
            const int cg = (ci << 4) + cl;
            v2f bw; bw[0] = nvalid * wlds[nw * CIN + cg + 0];
                    bw[1] = nvalid * wlds[nw * CIN + cg + 1];
            acc = __builtin_amdgcn_wmma_f32_16x16x4_f32(
                false, a, false, bw, (short)0, acc, false, false);
        }
        __syncthreads();                       // protect buffer reuse
    }

    // C layout: lane<16 -> rows 0..7, lane>=16 -> rows 8..15; column = m
    if (m < 3) {
        const float bias = conv_b[m] + out_bias[m];
        const int wcol = w0 + (wave << 4) + (khalf << 3);
        const size_t o0 = ((size_t)(b * CO + m) << 16) + ((size_t)h << 8) + wcol;
        float4 lo = make_float4(acc[0] + bias, acc[1] + bias, acc[2] + bias, acc[3] + bias);
        float4 hi = make_float4(acc[4] + bias, acc[5] + bias, acc[6] + bias, acc[7] + bias);
        *(float4*)(yout + o0)     = lo;
        *(float4*)(yout + o0 + 4) = hi;
    }
}

// =====================================================================
// K2: per-(b,o) mean / rstd over HW  -> stats[0..23]=mu, stats[24..47]=rstd
// =====================================================================
__global__ void __launch_bounds__(256)
reduce_stats(const float* __restrict__ y, float* __restrict__ stats)
{
    __shared__ float ssum[256], ssq[256];
    const int blk = blockIdx.x;                 // b*3+o, 0..23
    const float* p = y + ((size_t)blk << 16);
    float s = 0.f, q = 0.f;
    for (int i = threadIdx.x; i < HW; i += 256) { float v = p[i]; s += v; q += v * v; }
    ssum[threadIdx.x] = s; ssq[threadIdx.x] = q;
    __syncthreads();
    for (int off = 128; off > 0; off >>= 1) {
        if (threadIdx.x < off) {
            ssum[threadIdx.x] += ssum[threadIdx.x + off];
            ssq[threadIdx.x]  += ssq[threadIdx.x + off];
        }
        __syncthreads();
    }
    if (threadIdx.x == 0) {
        float mu  = ssum[0] * (1.0f / HW);
        float var = ssq[0] * (1.0f / HW) - mu * mu;
        stats[blk]      = mu;
        stats[24 + blk] = rsqrtf(var + 1e-5f);
    }
}

// =====================================================================
// K3: style linear  s[b][j] = <style_b, lin_w_j>/sqrt(512) + lin_b[j]
// =====================================================================
__global__ void style_mod(const float* __restrict__ style, const float* __restrict__ lin_w,
                          const float* __restrict__ lin_b, float* __restrict__ sbuf)
{
    const int t = threadIdx.x;
    if (t >= NB * 6) return;
    const int b = t / 6, j = t % 6;
    float acc = 0.f;
    for (int c = 0; c < CIN; ++c) acc += style[b * CIN + c] * lin_w[j * CIN + c];
    sbuf[t] = acc * INV_SQRT512 + lin_b[j];
}

// =====================================================================
// K4: normalize + style mod + upfirdn2d(up=2, pad=(2,1), [1,3,3,1]) add
// =====================================================================
__device__ __forceinline__ float skip_at(const float* sk, int y, int x) {
    return (y >= 0 && y < 128 && x >= 0 && x < 128) ? sk[y * 128 + x] : 0.0f;
}

__global__ void __launch_bounds__(256)
finalize(float* __restrict__ out, const float* __restrict__ skip,
         const float* __restrict__ stats, const float* __restrict__ sbuf)
{
    const int idx = blockIdx.x * 256 + threadIdx.x;
    if (idx >= NB * CO * HW) return;
    const int bo = idx >> 16;            // b*3+o
    const int p  = idx & (HW - 1);
    const int h  = p >> 8, w = p & 255;
    const int b  = bo / 3, o = bo % 3;

    const float mu = stats[bo], rs = stats[24 + bo];
    const float s0 = sbuf[b * 6 + o], s1 = sbuf[b * 6 + 3 + o];
    float r = (out[idx] - mu) * rs * (s0 + 1.0f) + s1;

    // separable 1D taps: even o -> 3/4*s[m] + 1/4*s[m-1]; odd o -> 1/4*s[m+1] + 3/4*s[m]
    int hy0, hy1, wx0, wx1; float fy0, fy1, fx0, fx1;
    if ((h & 1) == 0) { hy0 = h >> 1;       fy0 = 0.75f; hy1 = (h >> 1) - 1; fy1 = 0.25f; }
    else              { hy0 = (h + 1) >> 1; fy0 = 0.25f; hy1 = (h - 1) >> 1; fy1 = 0.75f; }
    if ((w & 1) == 0) { wx0 = w >> 1;       fx0 = 0.75f; wx1 = (w >> 1) - 1; fx1 = 0.25f; }
    else              { wx0 = (w + 1) >> 1; fx0 = 0.25f; wx1 = (w - 1) >> 1; fx1 = 0.75f; }

    const float* sk = skip + ((size_t)bo << 14);   // 128*128
    r += fy0 * (fx0 * skip_at(sk, hy0, wx0) + fx1 * skip_at(sk, hy0, wx1))
       + fy1 * (fx0 * skip_at(sk, hy1, wx0) + fx1 * skip_at(sk, hy1, wx1));

    out[idx] = r;
}

// =====================================================================
extern "C" void kernel_launch(void* const* d_in, const int* in_sizes, int n_in,
                              void* d_out, int out_size, void* d_ws, size_t ws_size,
                              hipStream_t stream)
{
    const float* x        = (const float*)d_in[0];
    const float* style    = (const float*)d_in[1];
    const float* skip     = (const float*)d_in[2];
    const float* conv_w   = (const float*)d_in[3];
    const float* conv_b   = (const float*)d_in[4];
    const float* out_bias = (const float*)d_in[5];
    const float* lin_w    = (const float*)d_in[6];
    const float* lin_b    = (const float*)d_in[7];

    float* out   = (float*)d_out;
    float* stats = (float*)d_ws;          // [0..23] mu, [24..47] rstd
    float* sbuf  = stats + 64;            // [48] style scale/shift

    const int nBlocksConv = NB * HW / PIX_BLK;      // 8192
    conv_wmma<<<nBlocksConv, 128, 0, stream>>>(x, conv_w, conv_b, out_bias, out);
    reduce_stats<<<NB * CO, 256, 0, stream>>>(out, stats);
    style_mod<<<1, 64, 0, stream>>>(style, lin_w, lin_b, sbuf);
    finalize<<<NB * CO * HW / 256, 256, 0, stream>>>(out, skip, stats, sbuf);
}